// NGCF_54047868453334
// MI455X (gfx1250) — compile-verified
//
#include <hip/hip_runtime.h>
#include <hip/hip_bf16.h>

// NGCF / 2-layer GCN for MI455X (gfx1250).
//
// Roofline: scatter-add over 4M edges x 64 f32 dominates (~2 GB/layer of
// random-ish traffic); feature matrices (38.4 MB) fit in the 192 MB L2, so
// the edge phase is L2-atomic bound. The per-layer GEMM (150000x64 @ 64x64,
// ~1.2 GFLOP) is negligible but is mapped onto V_WMMA_F32_16X16X4_F32 to use
// the CDNA5 matrix pipe at full f32 precision (bit-compatible accumulation
// class with the f32 reference).

#define DIM 64
#define NEG_SLOPE 0.01f

typedef float v2f __attribute__((ext_vector_type(2)));
typedef float v8f __attribute__((ext_vector_type(8)));

// ---------------------------------------------------------------- degrees ---
__global__ void ngcf_deg_kernel(const int* __restrict__ dst,
                                float* __restrict__ deg, int E) {
    int e = blockIdx.x * blockDim.x + threadIdx.x;
    if (e < E) atomicAdd(&deg[dst[e]], 1.0f);   // counts < 2^24 -> exact in f32
}

// deg -> rsqrt(deg + 1) in place (self-loop adds 1)
__global__ void ngcf_dinv_kernel(float* __restrict__ deg, int N) {
    int i = blockIdx.x * blockDim.x + threadIdx.x;
    if (i < N) deg[i] = rsqrtf(deg[i] + 1.0f);
}

// --------------------------------------------------------- WMMA f32 GEMM ---
// H[nrows x 64] = X[nrows x 64 (row stride ldx)] @ W[64 x 64]
// One wave per 16-row tile; 4 column tiles of 16; 16 k-steps of K=4.
// A 16x4 layout : lanes 0-15 -> K=kb+0/kb+1 ; lanes 16-31 -> K=kb+2/kb+3
// B 4x16 layout : vgpr0 = B[kb+khalf][n], vgpr1 = B[kb+khalf+1][n], n=lane&15
// D 16x16 layout: vgpr v -> row v (+8 for lanes 16-31), col = lane&15
__global__ void ngcf_gemm_wmma_kernel(const float* __restrict__ X, int ldx,
                                      const float* __restrict__ W,
                                      float* __restrict__ H, int nrows) {
    int wave = blockIdx.x * (blockDim.x >> 5) + (threadIdx.x >> 5);
    int row0 = wave << 4;
    if (row0 >= nrows) return;            // wave-uniform: EXEC stays all-1s
    int lane  = threadIdx.x & 31;
    int m     = lane & 15;
    int khalf = (lane >> 4) << 1;         // 0 or 2

    const float* xrow = X + (size_t)(row0 + m) * (size_t)ldx;
    v8f acc0 = {}, acc1 = {}, acc2 = {}, acc3 = {};

#pragma unroll
    for (int kk = 0; kk < 16; ++kk) {
        int kb = (kk << 2) + khalf;
        v2f a;
        a.x = xrow[kb];
        a.y = xrow[kb + 1];
        const float* w0 = W + kb * DIM + m;   // row kb, then row kb+1
        v2f b0 = { w0[0],  w0[DIM + 0]  };
        v2f b1 = { w0[16], w0[DIM + 16] };
        v2f b2 = { w0[32], w0[DIM + 32] };
        v2f b3 = { w0[48], w0[DIM + 48] };
        acc0 = __builtin_amdgcn_wmma_f32_16x16x4_f32(false, a, false, b0, (short)0, acc0, false, false);
        acc1 = __builtin_amdgcn_wmma_f32_16x16x4_f32(false, a, false, b1, (short)0, acc1, false, false);
        acc2 = __builtin_amdgcn_wmma_f32_16x16x4_f32(false, a, false, b2, (short)0, acc2, false, false);
        acc3 = __builtin_amdgcn_wmma_f32_16x16x4_f32(false, a, false, b3, (short)0, acc3, false, false);
    }

    int rbase = row0 + ((lane >> 4) << 3);
#pragma unroll
    for (int v = 0; v < 8; ++v) {
        float* hr = H + (size_t)(rbase + v) * DIM + m;
        hr[0]  = acc0[v];
        hr[16] = acc1[v];
        hr[32] = acc2[v];
        hr[48] = acc3[v];
    }
}

// ---------------------------------------------------------- edge scatter ---
// One wave per edge; lane handles float2 of the 64-wide row (coalesced b64
// load from h[src], two global_atomic_add_f32 into agg[dst], L2-resident).
__global__ void ngcf_scatter_kernel(const int* __restrict__ src,
                                    const int* __restrict__ dst,
                                    const float* __restrict__ dinv,
                                    const float* __restrict__ H,
                                    float* __restrict__ agg, int E) {
    unsigned long long tid = (unsigned long long)blockIdx.x * blockDim.x + threadIdx.x;
    int e = (int)(tid >> 5);
    if (e >= E) return;
    int lane = threadIdx.x & 31;
    int s = src[e];
    int d = dst[e];
    float norm = dinv[s] * dinv[d];
    float2 h2 = ((const float2*)(H + (size_t)s * DIM))[lane];
    float* ap = agg + (size_t)d * DIM + (lane << 1);
    atomicAdd(ap,     h2.x * norm);
    atomicAdd(ap + 1, h2.y * norm);
}

// ------------------------------------------------- self-loop + bias + act ---
// out[i,128] gets leaky_relu(agg + h*dinv^2 + b) at column offset col_off.
__global__ void ngcf_combine_kernel(const float* __restrict__ agg,
                                    const float* __restrict__ H,
                                    const float* __restrict__ dinv,
                                    const float* __restrict__ bias,
                                    float* __restrict__ out, int col_off, int N) {
    int tid = blockIdx.x * blockDim.x + threadIdx.x;   // N*64 < 2^31
    if (tid >= N * DIM) return;
    int i = tid >> 6;
    int d = tid & 63;
    float di = dinv[i];
    float v = agg[tid] + H[tid] * (di * di) + bias[d];
    v = (v > 0.0f) ? v : v * NEG_SLOPE;
    out[(size_t)i * (2 * DIM) + col_off + d] = v;
}

// -------------------------------------------------------------- launcher ---
extern "C" void kernel_launch(void* const* d_in, const int* in_sizes, int n_in,
                              void* d_out, int out_size, void* d_ws, size_t ws_size,
                              hipStream_t stream) {
    const int*   edge = (const int*)d_in[0];     // [2, E] int32
    const float* emb  = (const float*)d_in[1];   // [N, 64]
    const float* W1   = (const float*)d_in[2];   // [64, 64]
    const float* b1   = (const float*)d_in[3];   // [64]
    const float* W2   = (const float*)d_in[4];   // [64, 64]
    const float* b2   = (const float*)d_in[5];   // [64]

    const int E = in_sizes[0] / 2;
    const int N = in_sizes[1] / DIM;             // 150000, divisible by 16
    const int* src = edge;
    const int* dst = edge + E;

    float* ws   = (float*)d_ws;
    float* dinv = ws;                            // N floats (deg, then rsqrt)
    float* H    = ws + (size_t)N;                // N*64
    float* agg  = H + (size_t)N * DIM;           // N*64
    float* out  = (float*)d_out;                 // [N, 128]

    const int ntiles  = (N + 15) / 16;
    const int gblocks = (ntiles + 7) / 8;                       // 8 waves/block
    const int sblocks = (int)(((unsigned long long)E * 32 + 255) / 256);
    const int cblocks = (int)(((unsigned long long)N * DIM + 255) / 256);

    // degrees (shared across layers)
    hipMemsetAsync(dinv, 0, (size_t)N * sizeof(float), stream);
    ngcf_deg_kernel<<<(E + 255) / 256, 256, 0, stream>>>(dst, dinv, E);
    ngcf_dinv_kernel<<<(N + 255) / 256, 256, 0, stream>>>(dinv, N);

    // ---- layer 1: x1 = leaky(gcn(emb, W1, b1)) -> d_out cols [0,64)
    ngcf_gemm_wmma_kernel<<<gblocks, 256, 0, stream>>>(emb, DIM, W1, H, N);
    hipMemsetAsync(agg, 0, (size_t)N * DIM * sizeof(float), stream);
    ngcf_scatter_kernel<<<sblocks, 256, 0, stream>>>(src, dst, dinv, H, agg, E);
    ngcf_combine_kernel<<<cblocks, 256, 0, stream>>>(agg, H, dinv, b1, out, 0, N);

    // ---- layer 2: x2 = leaky(gcn(x1, W2, b2)) -> d_out cols [64,128)
    // layer-2 input read straight from d_out with row stride 128
    ngcf_gemm_wmma_kernel<<<gblocks, 256, 0, stream>>>(out, 2 * DIM, W2, H, N);
    hipMemsetAsync(agg, 0, (size_t)N * DIM * sizeof(float), stream);
    ngcf_scatter_kernel<<<sblocks, 256, 0, stream>>>(src, dst, dinv, H, agg, E);
    ngcf_combine_kernel<<<cblocks, 256, 0, stream>>>(agg, H, dinv, b2, out, DIM, N);
}